// Detector_26714696581167
// MI455X (gfx1250) — compile-verified
//
#include <hip/hip_runtime.h>
#include <hip/hip_bf16.h>
#include <math.h>

typedef __attribute__((ext_vector_type(2))) float v2f;
typedef __attribute__((ext_vector_type(8))) float v8f;

#define Bb   4
#define Ll   256
#define Hh   768
#define NCLS 9
#define H2   1536
#define NS   32896            // L*(L+1)/2
#define NEGV (-100000000.0f)

__device__ __forceinline__ float softplusf(float x) {
  return (x > 0.0f) ? (x + log1pf(expf(-x))) : log1pf(expf(x));
}

// ---------------------------------------------------------------------------
// Kernel A: labels softmax + per-query loc parameters (tiny GEMMs, wave/row).
// ---------------------------------------------------------------------------
__global__ void labels_loc_kernel(const float* __restrict__ queries,
                                  const float* __restrict__ queries_locs,
                                  const float* __restrict__ queries_logits,
                                  const float* __restrict__ W_cls,
                                  const float* __restrict__ b_cls,
                                  const float* __restrict__ W_loc,
                                  const float* __restrict__ b_loc,
                                  float* __restrict__ out_labels,
                                  float* __restrict__ locp) {
  const int wave = threadIdx.x >> 5;
  const int lane = threadIdx.x & 31;
  const int r = blockIdx.x * 8 + wave;           // row in [0, B*L)
  if (r >= Bb * Ll) return;

  float acc[13];
#pragma unroll
  for (int i = 0; i < 13; ++i) acc[i] = 0.0f;

  const float* q = queries + (size_t)r * Hh;
  for (int h = lane; h < Hh; h += 32) {
    const float qv = q[h];
    const float* wc = W_cls + h * NCLS;
#pragma unroll
    for (int t = 0; t < NCLS; ++t) acc[t] += qv * wc[t];
    const float* wl = W_loc + h * 4;
#pragma unroll
    for (int j = 0; j < 4; ++j) acc[NCLS + j] += qv * wl[j];
  }
#pragma unroll
  for (int i = 0; i < 13; ++i) {
#pragma unroll
    for (int off = 16; off > 0; off >>= 1)
      acc[i] += __shfl_xor(acc[i], off, 32);
  }

  if (lane == 0) {
    float lg[NCLS];
    float mx = -1e30f;
#pragma unroll
    for (int t = 0; t < NCLS; ++t) {
      lg[t] = queries_logits[r * NCLS + t] + acc[t] + b_cls[t];
      mx = fmaxf(mx, lg[t]);
    }
    float s = 0.0f;
#pragma unroll
    for (int t = 0; t < NCLS; ++t) { lg[t] = expf(lg[t] - mx); s += lg[t]; }
    const float inv = 1.0f / s;
#pragma unroll
    for (int t = 0; t < NCLS; ++t) out_labels[r * NCLS + t] = lg[t] * inv;

    const float l0 = acc[9]  + b_loc[0];
    const float l1 = acc[10] + b_loc[1];
    const float l2 = acc[11] + b_loc[2];
    const float l3 = acc[12] + b_loc[3];
    locp[r * 4 + 0] = queries_locs[r * 2 + 0] + l0;   // center start
    locp[r * 4 + 1] = queries_locs[r * 2 + 1] + l1;   // center end
    locp[r * 4 + 2] = softplusf(l2);                  // factor start
    locp[r * 4 + 3] = softplusf(l3);                  // factor end
  }
}

// ---------------------------------------------------------------------------
// Kernel B: projection GEMM  Out(1024x1536) = A(1024x768) * W(768x1536) + b.
// LDS-tiled, v_wmma_f32_16x16x4_f32. Block = 8 waves -> 128x64 output tile,
// wave = 16x64 (4 accumulators).
// ---------------------------------------------------------------------------
__global__ void gemm_proj_kernel(const float* __restrict__ A,
                                 const float* __restrict__ W,
                                 const float* __restrict__ bias,
                                 float* __restrict__ Out) {
  __shared__ float As[128 * 16];   // [row][k]   (row-major, row*16+k)
  __shared__ float Bs[64 * 18];    // [n][k]     (pad 18 keeps 8B alignment)

  const int tid  = threadIdx.x;
  const int w    = tid >> 5;
  const int lane = tid & 31;
  const int lo   = lane & 15;
  const int half = lane >> 4;
  const int mblock = blockIdx.y * 128;
  const int nblock = blockIdx.x * 64;

  v8f acc[4] = {};

  for (int kb = 0; kb < Hh; kb += 16) {
    __syncthreads();
    // A tile: 128x16 = 2048 floats, 8 per thread, coalesced in k.
#pragma unroll
    for (int i = 0; i < 8; ++i) {
      const int idx = tid + i * 256;
      const int row = idx >> 4, k = idx & 15;
      As[idx] = A[(size_t)(mblock + row) * Hh + kb + k];
    }
    // B tile: 16x64 = 1024 floats, 4 per thread, coalesced in n; store [n][k].
#pragma unroll
    for (int i = 0; i < 4; ++i) {
      const int idx = tid + i * 256;
      const int k = idx >> 6, n = idx & 63;
      Bs[n * 18 + k] = W[(size_t)(kb + k) * H2 + nblock + n];
    }
    __syncthreads();

#pragma unroll
    for (int step = 0; step < 4; ++step) {
      const v2f a = *(const v2f*)&As[(w * 16 + lo) * 16 + step * 4 + 2 * half];
#pragma unroll
      for (int nt = 0; nt < 4; ++nt) {
        const v2f bb = *(const v2f*)&Bs[(nt * 16 + lo) * 18 + step * 4 + 2 * half];
        acc[nt] = __builtin_amdgcn_wmma_f32_16x16x4_f32(
            false, a, false, bb, (short)0, acc[nt], false, false);
      }
    }
  }

#pragma unroll
  for (int nt = 0; nt < 4; ++nt) {
#pragma unroll
    for (int v = 0; v < 8; ++v) {
      const int row = mblock + w * 16 + v + 8 * half;
      const int col = nblock + nt * 16 + lo;
      Out[(size_t)row * H2 + col] = acc[nt][v] + bias[col];
    }
  }
}

// ---------------------------------------------------------------------------
// Kernel C: score[b,c,q,v] = <qh[b,q,c,:], ch[b,v,c,:]> / sqrt(H), masked.
// One wave per 16x16 tile; B operand = ch^T, whose k-pairs are contiguous
// in memory -> clean b64 loads matching the WMMA B layout.
// ---------------------------------------------------------------------------
__global__ void score_kernel(const float* __restrict__ qh,
                             const float* __restrict__ ch,
                             const unsigned char* __restrict__ masks,
                             float* __restrict__ score) {
  const int lane = threadIdx.x & 31;
  const int lo = lane & 15, half = lane >> 4;
  const int m0 = blockIdx.y * 16;
  const int n0 = blockIdx.x * 16;
  const int bc = blockIdx.z;                 // b*2 + c
  const int b = bc >> 1, c = bc & 1;
  const float rfac = 0.03608439182435161f;   // 1/sqrt(768)

  const float* Abase = qh + (size_t)(b * Ll + m0 + lo) * H2 + c * Hh;
  const float* Bbase = ch + (size_t)(b * Ll + n0 + lo) * H2 + c * Hh;

  v8f acc = {};
  for (int k = 0; k < Hh; k += 4) {
    const v2f a  = *(const v2f*)(Abase + k + 2 * half);
    const v2f bb = *(const v2f*)(Bbase + k + 2 * half);
    acc = __builtin_amdgcn_wmma_f32_16x16x4_f32(
        false, a, false, bb, (short)0, acc, false, false);
  }

  const int n = n0 + lo;
  const bool ok = masks[b * Ll + n] != 0;
#pragma unroll
  for (int v = 0; v < 8; ++v) {
    const int m = m0 + v + 8 * half;
    score[((size_t)bc * Ll + m) * Ll + n] = ok ? acc[v] * rfac : NEGV;
  }
}

// ---------------------------------------------------------------------------
// Kernel D: span softmax. One block per (b,q) row; score rows staged in LDS;
// online softmax pass + single-write recompute pass (134.7MB written once).
// ---------------------------------------------------------------------------
__global__ void span_softmax_kernel(const float* __restrict__ score,
                                    const float* __restrict__ locp,
                                    float* __restrict__ out_locs) {
  __shared__ float s0[Ll], s1[Ll];
  __shared__ float rm[256], rs[256];

  const int t = threadIdx.x;
  const int r = blockIdx.x;          // b*L + q
  const int b = r >> 8, q = r & 255;

  s0[t] = score[((size_t)(b * 2 + 0) * Ll + q) * Ll + t];
  s1[t] = score[((size_t)(b * 2 + 1) * Ll + q) * Ll + t];
  const float loc0 = locp[r * 4 + 0], loc1 = locp[r * 4 + 1];
  const float f0   = locp[r * 4 + 2], f1   = locp[r * 4 + 3];
  __syncthreads();

  // Pass 1: online max/sum over spans p = t, t+256, ...
  float m = -INFINITY, ssum = 0.0f;
  for (int p = t; p < NS; p += 256) {
    int e = (int)((sqrtf(8.0f * (float)p + 1.0f) - 1.0f) * 0.5f);
    while ((e + 1) * (e + 2) / 2 <= p) ++e;
    while (e * (e + 1) / 2 > p) --e;
    const int s = p - e * (e + 1) / 2;
    const float ds = (float)s - loc0, de = (float)e - loc1;
    const float val = s0[s] + s1[e] - (ds * ds * f0 + de * de * f1);
    if (val > m) { ssum = ssum * expf(m - val) + 1.0f; m = val; }
    else          ssum += expf(val - m);
  }
  rm[t] = m; rs[t] = ssum;
  __syncthreads();
  for (int off = 128; off > 0; off >>= 1) {
    if (t < off) {
      const float m2 = rm[t + off], c2 = rs[t + off];
      const float M = fmaxf(rm[t], m2);
      rs[t] = rs[t] * expf(rm[t] - M) + c2 * expf(m2 - M);
      rm[t] = M;
    }
    __syncthreads();
  }
  const float M = rm[0];
  const float invS = 1.0f / rs[0];

  // Pass 2: recompute and write each output exactly once.
  float* outp = out_locs + (size_t)r * NS;
  for (int p = t; p < NS; p += 256) {
    int e = (int)((sqrtf(8.0f * (float)p + 1.0f) - 1.0f) * 0.5f);
    while ((e + 1) * (e + 2) / 2 <= p) ++e;
    while (e * (e + 1) / 2 > p) --e;
    const int s = p - e * (e + 1) / 2;
    const float ds = (float)s - loc0, de = (float)e - loc1;
    const float val = s0[s] + s1[e] - (ds * ds * f0 + de * de * f1);
    outp[p] = expf(val - M) * invS;
  }
}

// ---------------------------------------------------------------------------
extern "C" void kernel_launch(void* const* d_in, const int* in_sizes, int n_in,
                              void* d_out, int out_size, void* d_ws, size_t ws_size,
                              hipStream_t stream) {
  (void)in_sizes; (void)n_in; (void)out_size; (void)ws_size;

  const float* batch_hiddens        = (const float*)d_in[0];
  const unsigned char* batch_masks  = (const unsigned char*)d_in[1];
  const float* queries              = (const float*)d_in[2];
  const float* queries_locs         = (const float*)d_in[3];
  const float* queries_logits       = (const float*)d_in[4];
  const float* W_cls                = (const float*)d_in[5];
  const float* b_cls                = (const float*)d_in[6];
  const float* W_q                  = (const float*)d_in[7];
  const float* b_q                  = (const float*)d_in[8];
  const float* W_ctx                = (const float*)d_in[9];
  const float* b_ctx                = (const float*)d_in[10];
  const float* W_loc                = (const float*)d_in[11];
  const float* b_loc                = (const float*)d_in[12];

  float* out        = (float*)d_out;
  float* out_labels = out;
  float* out_locs   = out + (size_t)Bb * Ll * NCLS;

  float* ws    = (float*)d_ws;
  float* qh    = ws;                                  // 1024*1536
  float* ch    = qh + (size_t)Bb * Ll * H2;           // 1024*1536
  float* score = ch + (size_t)Bb * Ll * H2;           // 8*256*256
  float* locp  = score + (size_t)Bb * 2 * Ll * Ll;    // 1024*4

  labels_loc_kernel<<<dim3(128), dim3(256), 0, stream>>>(
      queries, queries_locs, queries_logits, W_cls, b_cls, W_loc, b_loc,
      out_labels, locp);

  gemm_proj_kernel<<<dim3(H2 / 64, (Bb * Ll) / 128), dim3(256), 0, stream>>>(
      queries, W_q, b_q, qh);
  gemm_proj_kernel<<<dim3(H2 / 64, (Bb * Ll) / 128), dim3(256), 0, stream>>>(
      batch_hiddens, W_ctx, b_ctx, ch);

  score_kernel<<<dim3(16, 16, Bb * 2), dim3(32), 0, stream>>>(
      qh, ch, batch_masks, score);

  span_softmax_kernel<<<dim3(Bb * Ll), dim3(256), 0, stream>>>(
      score, locp, out_locs);
}